// GraphItem2Vec_1735166787760
// MI455X (gfx1250) — compile-verified
//
#include <hip/hip_runtime.h>
#include <hip/hip_bf16.h>

typedef __attribute__((ext_vector_type(2))) float v2f;
typedef __attribute__((ext_vector_type(8))) float v8f;

// ---------------------------------------------------------------------------
// Fill a float buffer with a constant value.
// ---------------------------------------------------------------------------
__global__ void __launch_bounds__(256) k_fill(float* __restrict__ p, float v, size_t n) {
    size_t i = (size_t)blockIdx.x * blockDim.x + threadIdx.x;
    if (i < n) p[i] = v;
}

// ---------------------------------------------------------------------------
// Degree counting: deg[src[e]] += 1, cnt[dst[e]] += 1  (self-loops folded into
// the init value 1.0f).
// ---------------------------------------------------------------------------
__global__ void __launch_bounds__(256) k_count(const int* __restrict__ esrc,
                                               const int* __restrict__ edst,
                                               float* __restrict__ deg,
                                               float* __restrict__ cnt, int E) {
    int e = blockIdx.x * blockDim.x + threadIdx.x;
    if (e < E) {
        atomicAdd(&deg[esrc[e]], 1.0f);
        atomicAdd(&cnt[edst[e]], 1.0f);
    }
}

// ---------------------------------------------------------------------------
// In-place: deg -> deg^-0.5 (dis), cnt -> 1/cnt (invcnt).
// ---------------------------------------------------------------------------
__global__ void __launch_bounds__(256) k_recip(float* __restrict__ deg_to_dis,
                                               float* __restrict__ cnt_to_inv, int N) {
    int i = blockIdx.x * blockDim.x + threadIdx.x;
    if (i < N) {
        deg_to_dis[i] = 1.0f / sqrtf(deg_to_dis[i]);
        cnt_to_inv[i] = 1.0f / cnt_to_inv[i];
    }
}

// ---------------------------------------------------------------------------
// Edge-parallel scatter: one wave32 per edge, 4 floats per lane (128 dims).
//   s[dst] += dis[src]*dis[dst] * x[src]
// Edges [0,E) come from edge_index; edges [E, E+N) are self-loops.
// ---------------------------------------------------------------------------
__global__ void __launch_bounds__(256) k_scatter(const float* __restrict__ x,
                                                 float* __restrict__ s,
                                                 const int* __restrict__ esrc,
                                                 const int* __restrict__ edst,
                                                 const float* __restrict__ dis,
                                                 int E, int N) {
    int gid  = blockIdx.x * blockDim.x + threadIdx.x;
    int edge = gid >> 5;
    int lane = gid & 31;
    if (edge >= E + N) return;
    int si, di;
    if (edge < E) { si = esrc[edge]; di = edst[edge]; }
    else          { si = edge - E;   di = si; }
    float nrm = dis[si] * dis[di];
    float4 v = *((const float4*)(x + (size_t)si * 128) + lane);
    float* so = s + (size_t)di * 128 + (size_t)lane * 4;
    atomicAdd(so + 0, nrm * v.x);
    atomicAdd(so + 1, nrm * v.y);
    atomicAdd(so + 2, nrm * v.z);
    atomicAdd(so + 3, nrm * v.w);
}

// ---------------------------------------------------------------------------
// x1 = s * invcnt  (mean over incoming edges); reset s to 0 for next layer.
// float4 per thread, 32 threads per node row.
// ---------------------------------------------------------------------------
__global__ void __launch_bounds__(256) k_norm_reset(float* __restrict__ s,
                                                    const float* __restrict__ invcnt,
                                                    float* __restrict__ x1, int N) {
    int gid  = blockIdx.x * blockDim.x + threadIdx.x;
    int node = gid >> 5;
    int q    = gid & 31;
    if (node >= N) return;
    float ic = invcnt[node];
    float4* sp = (float4*)(s + (size_t)node * 128) + q;
    float4 v = *sp;
    float4 r; r.x = v.x * ic; r.y = v.y * ic; r.z = v.z * ic; r.w = v.w * ic;
    ((float4*)(x1 + (size_t)node * 128))[q] = r;
    float4 z; z.x = 0.f; z.y = 0.f; z.z = 0.f; z.w = 0.f;
    *sp = z;
}

// ---------------------------------------------------------------------------
// eout = (emb + x1 + s*invcnt) / 3
// ---------------------------------------------------------------------------
__global__ void __launch_bounds__(256) k_final(const float* __restrict__ emb,
                                               const float* __restrict__ x1,
                                               const float* __restrict__ s,
                                               const float* __restrict__ invcnt,
                                               float* __restrict__ eout, int N) {
    int gid  = blockIdx.x * blockDim.x + threadIdx.x;
    int node = gid >> 5;
    int q    = gid & 31;
    if (node >= N) return;
    float ic = invcnt[node];
    const float4 e  = ((const float4*)(emb + (size_t)node * 128))[q];
    const float4 a  = ((const float4*)(x1  + (size_t)node * 128))[q];
    const float4 b  = ((const float4*)(s   + (size_t)node * 128))[q];
    const float third = 1.0f / 3.0f;
    float4 r;
    r.x = (e.x + a.x + b.x * ic) * third;
    r.y = (e.y + a.y + b.y * ic) * third;
    r.z = (e.z + a.z + b.z * ic) * third;
    r.w = (e.w + a.w + b.w * ic) * third;
    ((float4*)(eout + (size_t)node * 128))[q] = r;
}

// ---------------------------------------------------------------------------
// Scoring with V_WMMA_F32_16X16X4_F32 (fp32 A/B, exact math).
// One wave per (batch b, tile of 16 samples).
//   A[m,k] = eout[samples[b, m0+m], k]   (16 x 4 fragment per step)
//   B[k,n] = eout[items[b], k]           (item vector broadcast to all 16 cols)
//   C[m,0] = score(b, m0+m)              (extract column N=0)
// A fragment layout (ISA 7.12.2): lanes 0-15 -> M=0..15, VGPR0/1 = K,K+1;
// lanes 16-31 -> M=0..15, VGPR0/1 = K+2,K+3.  B values depend only on K, so
// the broadcast-item fragment uses the same K offsets.
// C layout: column 0 lives in lane 0 (M=0..7 in c[0..7]) and lane 16 (M=8..15).
// ---------------------------------------------------------------------------
__global__ void __launch_bounds__(256) k_score(const float* __restrict__ eout,
                                               const int* __restrict__ items,
                                               const int* __restrict__ samples,
                                               float* __restrict__ out,
                                               int B, int S) {
    int gid  = blockIdx.x * blockDim.x + threadIdx.x;
    int wave = gid >> 5;
    int lane = threadIdx.x & 31;
    int tilesPerB = (S + 15) >> 4;
    int b    = wave / tilesPerB;
    int tile = wave - b * tilesPerB;
    if (b >= B) return;

    int m0 = tile << 4;
    int ml = m0 + (lane & 15);
    int msafe = (ml < S) ? ml : 0;                      // clamp; extra rows discarded
    int sidx = samples[(size_t)b * S + msafe];
    int iidx = items[b];
    const float* rowA = eout + (size_t)sidx * 128;
    const float* rowB = eout + (size_t)iidx * 128;
    int khalf = (lane >> 4) << 1;                       // 0 for lanes 0-15, 2 for 16-31

    v8f c = {};
#pragma unroll
    for (int k = 0; k < 128; k += 4) {
        v2f a  = *(const v2f*)(rowA + k + khalf);
        v2f bb = *(const v2f*)(rowB + k + khalf);
        c = __builtin_amdgcn_wmma_f32_16x16x4_f32(
                /*neg_a=*/false, a, /*neg_b=*/false, bb,
                /*c_mod=*/(short)0, c, /*reuse_a=*/false, /*reuse_b=*/false);
    }

    // Extract column N=0 of the 16x16 result.
    if ((lane & 15) == 0) {
        int mbase = m0 + ((lane >> 4) << 3);            // lane 0 -> m0, lane 16 -> m0+8
#pragma unroll
        for (int r = 0; r < 8; ++r) {
            int m = mbase + r;
            if (m < S) out[(size_t)b * S + m] = c[r];
        }
    }
}

// ---------------------------------------------------------------------------
// Host-side orchestration (graph-capture safe: only kernel launches on stream).
// ---------------------------------------------------------------------------
extern "C" void kernel_launch(void* const* d_in, const int* in_sizes, int n_in,
                              void* d_out, int out_size, void* d_ws, size_t ws_size,
                              hipStream_t stream) {
    const float* emb        = (const float*)d_in[0];
    const int*   edge_index = (const int*)d_in[1];
    const int*   items      = (const int*)d_in[2];
    const int*   samples    = (const int*)d_in[3];
    float*       out        = (float*)d_out;

    const int N = in_sizes[0] / 128;        // 100000 nodes
    const int E = in_sizes[1] / 2;          // 600000 edges
    const int B = in_sizes[2];              // 4096 items
    const int S = in_sizes[3] / B;          // 100 samples per item

    const int* esrc = edge_index;           // edge_index[0, :]
    const int* edst = edge_index + E;       // edge_index[1, :]

    // Workspace layout (floats): S | X1 | EOUT | DEG | CNT  ~= 154.4 MB
    float* Sbuf = (float*)d_ws;
    float* X1   = Sbuf + (size_t)N * 128;
    float* EOUT = X1   + (size_t)N * 128;
    float* DEG  = EOUT + (size_t)N * 128;
    float* CNT  = DEG  + N;

    const int T = 256;
    auto blocks = [](size_t work, int t) { return (unsigned)((work + t - 1) / t); };

    // Init: scatter accumulator = 0; deg/cnt = 1 (self-loop contribution).
    k_fill<<<blocks((size_t)N * 128, T), T, 0, stream>>>(Sbuf, 0.0f, (size_t)N * 128);
    k_fill<<<blocks((size_t)N, T), T, 0, stream>>>(DEG, 1.0f, (size_t)N);
    k_fill<<<blocks((size_t)N, T), T, 0, stream>>>(CNT, 1.0f, (size_t)N);

    // Degrees, then dis = deg^-0.5 and invcnt = 1/cnt (in place).
    k_count<<<blocks((size_t)E, T), T, 0, stream>>>(esrc, edst, DEG, CNT, E);
    k_recip<<<blocks((size_t)N, T), T, 0, stream>>>(DEG, CNT, N);

    // Layer 1: S += norm * emb[src];  X1 = S/cnt;  S = 0.
    size_t scatterThreads = (size_t)(E + N) * 32;
    k_scatter<<<blocks(scatterThreads, T), T, 0, stream>>>(emb, Sbuf, esrc, edst, DEG, E, N);
    k_norm_reset<<<blocks((size_t)N * 32, T), T, 0, stream>>>(Sbuf, CNT, X1, N);

    // Layer 2: S += norm * X1[src];  EOUT = (emb + X1 + S/cnt)/3.
    k_scatter<<<blocks(scatterThreads, T), T, 0, stream>>>(X1, Sbuf, esrc, edst, DEG, E, N);
    k_final<<<blocks((size_t)N * 32, T), T, 0, stream>>>(emb, X1, Sbuf, CNT, EOUT, N);

    // Scoring with fp32 WMMA: one wave per (b, 16-sample tile).
    int tilesPerB = (S + 15) / 16;
    size_t scoreThreads = (size_t)B * tilesPerB * 32;
    k_score<<<blocks(scoreThreads, T), T, 0, stream>>>(EOUT, items, samples, out, B, S);
}